// STDecoupled_KD_70042326663998
// MI455X (gfx1250) — compile-verified
//
#include <hip/hip_runtime.h>
#include <hip/hip_bf16.h>
#include <math.h>

// ---------------------------------------------------------------------------
// ST-Decoupled KD loss for MI455X (gfx1250, wave32).
//  - 1 workgroup (512 thr = 16 wave32) per row; row data (~600KB) stays in L2.
//  - Pass 1: b128 streaming; maxes of t / s_score / unif (gumbel is monotone
//    so max(s) <= max(s_score)+gumbel(max(unif)) -- no logs in this pass) and
//    the level-1 (11-bit) histogram of descending-sortable teacher keys.
//  - Pass 2: softmax denominators via V_WMMA_F32_16X16X4_F32 (A x ones + C),
//    4 independent accumulation chains, branch-free main loop + guarded tail.
//    exp(s+g-M) = exp(s-M) * rcp(w), w = -log(u+eps)+eps  (v_rcp_f32, no IEEE
//    div sequence; hardware __logf/__expf in the bulk pass).
//  - Rank select (ranks 0..9 and 10+neg_cols): 2-level MSD radix select with
//    per-needed-bin 7-bit level-2 histograms and tiny LDS candidate lists.
//  - Thread 0 finishes the 10/50-way CE terms; per-row loss -> d_ws; second
//    kernel reduces the 512 partials deterministically into d_out[0].
// ---------------------------------------------------------------------------

#define B_ROWS 512
#define V_COLS 50257
#define NPOS   10
#define NNEG   50
#define NRANK  60            // 10 pos ranks + 50 neg ranks
#define NBIN   2048          // level-1 bins: key bits [31:21]
#define NSUB   128           // level-2 bins: key bits [20:14]
#define CAP    96            // candidate list capacity per needed (bin,sub)
#define BLOCK  512
#define EPSF   1e-20f

typedef __attribute__((ext_vector_type(2))) float v2f;
typedef __attribute__((ext_vector_type(8))) float v8f;

// Descending-sortable key: larger float -> smaller key.
__device__ __forceinline__ unsigned dkey(float f) {
  unsigned u = __float_as_uint(f);
  unsigned m = u ^ ((u & 0x80000000u) ? 0xFFFFFFFFu : 0x80000000u); // ascending
  return ~m;                                                        // descending
}

// Precise version (used only on the 60 gathered elements per row).
__device__ __forceinline__ float gumbel_of(float u) {
  return -logf(-logf(u + EPSF) + EPSF);
}

// Bulk-pass fast helpers: v_log_f32 / v_exp_f32 / v_rcp_f32 directly.
__device__ __forceinline__ float fexp(float x) { return __expf(x); }
__device__ __forceinline__ float flog(float x) { return __logf(x); }
__device__ __forceinline__ float frcp(float x) { return __builtin_amdgcn_rcpf(x); }

// op: 0 = sum, 1 = max. Block-uniform call; result broadcast to all threads.
__device__ __forceinline__ float blockRed(float v, float* red, int tid, int op) {
  __syncthreads();
  for (int o = 16; o > 0; o >>= 1) {
    float w = __shfl_xor(v, o, 32);
    v = op ? fmaxf(v, w) : (v + w);
  }
  if ((tid & 31) == 0) red[tid >> 5] = v;
  __syncthreads();
  if (tid == 0) {
    float a = red[0];
    for (int i = 1; i < 16; i++) a = op ? fmaxf(a, red[i]) : (a + red[i]);
    red[0] = a;
  }
  __syncthreads();
  return red[0];
}

__device__ __forceinline__ int bfind(const unsigned* a, int n, unsigned key) {
  int lo = 0, hi = n - 1;
  while (lo <= hi) {
    int mid = (lo + hi) >> 1;
    unsigned v = a[mid];
    if (v == key) return mid;
    if (v < key) lo = mid + 1; else hi = mid - 1;
  }
  return -1;
}

__global__ void __launch_bounds__(BLOCK)
kd_main(const float* __restrict__ t_score, const float* __restrict__ s_score,
        const int* __restrict__ gt, const float* __restrict__ unif,
        const int* __restrict__ neg_cols, float* __restrict__ row_loss) {
  __shared__ unsigned sh_hp[NBIN + 1];               // hist -> excl prefix
  __shared__ unsigned long long sh_buf[NRANK * CAP]; // hist2 region, reused as lists
  __shared__ float sh_red[16];
  __shared__ float sh_tgt, sh_sgt;
  __shared__ unsigned rk_bin[NRANK], rk_res1[NRANK], rk_res2[NRANK], rk_pkey[NRANK];
  __shared__ int rk_q[NRANK], rk_pslot[NRANK], sel_idx[NRANK];
  __shared__ unsigned dlist[NRANK], plist[NRANK], cursors[NRANK];
  __shared__ int s_nd, s_np;

  const int tid = threadIdx.x;
  const int row = blockIdx.x;
  const float* t_row = t_score + (size_t)row * V_COLS;
  const float* s_row = s_score + (size_t)row * V_COLS;
  const float* u_row = unif + (size_t)row * V_COLS;
  const int g = gt[row];

  for (int i = tid; i < NBIN + 1; i += BLOCK) sh_hp[i] = 0;
  __syncthreads();

  // ---------------- Pass 1: maxes (no transcendentals) + level-1 histogram --
  // Row base is only 4B-aligned (V odd): peel to 16B alignment, then b128.
  float mt = -INFINITY, msc = -INFINITY, mu = -INFINITY;
  {
    const int peel = (4 - (row & 3)) & 3; // (row + peel) % 4 == 0
    auto do1 = [&](int c, float t, float sv, float uv) {
      mt = fmaxf(mt, t); msc = fmaxf(msc, sv); mu = fmaxf(mu, uv);
      if (c != g) atomicAdd(&sh_hp[dkey(t) >> 21], 1u);
    };
    if (tid < peel) do1(tid, t_row[tid], s_row[tid], u_row[tid]);
    const int n4 = (V_COLS - peel) >> 2;
    for (int q = tid; q < n4; q += BLOCK) {
      const int c = peel + (q << 2);
      __builtin_prefetch(t_row + c + 4 * BLOCK, 0, 1);
      __builtin_prefetch(s_row + c + 4 * BLOCK, 0, 1);
      __builtin_prefetch(u_row + c + 4 * BLOCK, 0, 1);
      float4 tv = *(const float4*)(t_row + c);
      float4 sv = *(const float4*)(s_row + c);
      float4 uv = *(const float4*)(u_row + c);
      do1(c + 0, tv.x, sv.x, uv.x);
      do1(c + 1, tv.y, sv.y, uv.y);
      do1(c + 2, tv.z, sv.z, uv.z);
      do1(c + 3, tv.w, sv.w, uv.w);
    }
    for (int c = peel + (n4 << 2) + tid; c < V_COLS; c += BLOCK)
      do1(c, t_row[c], s_row[c], u_row[c]);
  }
  const float maxt = blockRed(mt, sh_red, tid, 1);
  const float maxsc = blockRed(msc, sh_red, tid, 1);
  const float maxu = blockRed(mu, sh_red, tid, 1);
  const float maxs = maxsc + gumbel_of(maxu); // upper bound on max(s+gumbel)
  if (tid == 0) {
    sh_tgt = t_row[g];
    sh_sgt = s_row[g] + gumbel_of(u_row[g]);
  }

  // ---------------- Pass 2: softmax denominators via WMMA ------------------
  // Each wave feeds 64 exp() values/instr as the 16x4 f32 A-matrix, B = ones:
  // D[m][n] = sum_k A[m][k] + C  => sum(all D)/16 == sum(all A fed).
  float sumt, sums;
  {
    v8f a0t = {0.f,0.f,0.f,0.f,0.f,0.f,0.f,0.f};
    v8f a0s = {0.f,0.f,0.f,0.f,0.f,0.f,0.f,0.f};
    v8f a1t = {0.f,0.f,0.f,0.f,0.f,0.f,0.f,0.f};
    v8f a1s = {0.f,0.f,0.f,0.f,0.f,0.f,0.f,0.f};
    v2f ones; ones.x = 1.0f; ones.y = 1.0f;

    // 24 branch-free iterations: 2048 cols each, 4 independent WMMA chains.
    for (int i = 0; i < 24; ++i) {
      const int c0 = i * 2048 + tid * 2;   // chain 0: cols [i*2048, i*2048+1023]
      const int c1 = c0 + 1024;            // chain 1
      __builtin_prefetch(t_row + c0 + 2048, 0, 1);
      __builtin_prefetch(s_row + c0 + 2048, 0, 1);
      __builtin_prefetch(u_row + c0 + 2048, 0, 1);
      float r00 = frcp(-flog(u_row[c0] + EPSF) + EPSF);
      float r01 = frcp(-flog(u_row[c0 + 1] + EPSF) + EPSF);
      float r10 = frcp(-flog(u_row[c1] + EPSF) + EPSF);
      float r11 = frcp(-flog(u_row[c1 + 1] + EPSF) + EPSF);
      v2f at0, as0, at1, as1;
      at0.x = fexp(t_row[c0] - maxt);       at0.y = fexp(t_row[c0 + 1] - maxt);
      at1.x = fexp(t_row[c1] - maxt);       at1.y = fexp(t_row[c1 + 1] - maxt);
      as0.x = fexp(s_row[c0] - maxs) * r00; as0.y = fexp(s_row[c0 + 1] - maxs) * r01;
      as1.x = fexp(s_row[c1] - maxs) * r10; as1.y = fexp(s_row[c1 + 1] - maxs) * r11;
      a0t = __builtin_amdgcn_wmma_f32_16x16x4_f32(false, at0, false, ones, (short)0, a0t, false, false);
      a0s = __builtin_amdgcn_wmma_f32_16x16x4_f32(false, as0, false, ones, (short)0, a0s, false, false);
      a1t = __builtin_amdgcn_wmma_f32_16x16x4_f32(false, at1, false, ones, (short)0, a1t, false, false);
      a1s = __builtin_amdgcn_wmma_f32_16x16x4_f32(false, as1, false, ones, (short)0, a1s, false, false);
    }
    // Tail: chain 0 covers 49152..50175 (in bounds), chain 1 is guarded.
    {
      const int c0 = 24 * 2048 + tid * 2;
      const int c1 = c0 + 1024;
      float r00 = frcp(-flog(u_row[c0] + EPSF) + EPSF);
      float r01 = frcp(-flog(u_row[c0 + 1] + EPSF) + EPSF);
      v2f at0, as0, at1, as1;
      at0.x = fexp(t_row[c0] - maxt);       at0.y = fexp(t_row[c0 + 1] - maxt);
      as0.x = fexp(s_row[c0] - maxs) * r00; as0.y = fexp(s_row[c0 + 1] - maxs) * r01;
      at1.x = 0.f; at1.y = 0.f; as1.x = 0.f; as1.y = 0.f;
      if (c1 < V_COLS) {
        float r10 = frcp(-flog(u_row[c1] + EPSF) + EPSF);
        at1.x = fexp(t_row[c1] - maxt);
        as1.x = fexp(s_row[c1] - maxs) * r10;
      }
      if (c1 + 1 < V_COLS) {
        float r11 = frcp(-flog(u_row[c1 + 1] + EPSF) + EPSF);
        at1.y = fexp(t_row[c1 + 1] - maxt);
        as1.y = fexp(s_row[c1 + 1] - maxs) * r11;
      }
      a0t = __builtin_amdgcn_wmma_f32_16x16x4_f32(false, at0, false, ones, (short)0, a0t, false, false);
      a0s = __builtin_amdgcn_wmma_f32_16x16x4_f32(false, as0, false, ones, (short)0, a0s, false, false);
      a1t = __builtin_amdgcn_wmma_f32_16x16x4_f32(false, at1, false, ones, (short)0, a1t, false, false);
      a1s = __builtin_amdgcn_wmma_f32_16x16x4_f32(false, as1, false, ones, (short)0, a1s, false, false);
    }
    float pt = 0.f, ps = 0.f;
    for (int i = 0; i < 8; i++) { pt += a0t[i] + a1t[i]; ps += a0s[i] + a1s[i]; }
    sumt = blockRed(pt, sh_red, tid, 0) * (1.0f / 16.0f);
    sums = blockRed(ps, sh_red, tid, 0) * (1.0f / 16.0f);
  }

  // ---------------- Rank machinery -----------------------------------------
  if (tid == 0) { // exclusive prefix of level-1 histogram
    unsigned acc = 0;
    for (int b = 0; b < NBIN; b++) { unsigned h = sh_hp[b]; sh_hp[b] = acc; acc += h; }
    sh_hp[NBIN] = acc; // = V-1 (gt excluded)
  }
  __syncthreads();

  // R1: each rank -> level-1 bin + residual
  if (tid < NRANK) {
    unsigned r = (tid < NPOS) ? (unsigned)tid
                              : (unsigned)(NPOS + neg_cols[row * NNEG + (tid - NPOS)]);
    int lo = 0, hi = NBIN - 1;
    while (lo < hi) { int mid = (lo + hi + 1) >> 1; if (sh_hp[mid] <= r) lo = mid; else hi = mid - 1; }
    rk_bin[tid] = (unsigned)lo;
    rk_res1[tid] = r - sh_hp[lo];
  }
  __syncthreads();
  if (tid == 0) { // dedupe + sort needed bins
    int nd = 0;
    for (int j = 0; j < NRANK; j++) {
      unsigned b = rk_bin[j]; bool f = false;
      for (int k = 0; k < nd; k++) if (dlist[k] == b) { f = true; break; }
      if (!f) dlist[nd++] = b;
    }
    for (int a = 1; a < nd; a++) {
      unsigned x = dlist[a]; int k = a - 1;
      while (k >= 0 && dlist[k] > x) { dlist[k + 1] = dlist[k]; k--; }
      dlist[k + 1] = x;
    }
    s_nd = nd;
  }
  __syncthreads();
  if (tid < NRANK) rk_q[tid] = bfind(dlist, s_nd, rk_bin[tid]);

  // R2: level-2 histograms for needed bins only (one streaming pass over t)
  unsigned* hist2 = (unsigned*)sh_buf; // NRANK*NSUB counters inside sh_buf
  for (int i = tid; i < NRANK * NSUB; i += BLOCK) hist2[i] = 0;
  __syncthreads();
  {
    const int nd = s_nd;
    for (int c = tid; c < V_COLS; c += BLOCK) {
      if (c == g) continue;
      unsigned d = dkey(t_row[c]);
      int q = bfind(dlist, nd, d >> 21);
      if (q >= 0) atomicAdd(&hist2[q * NSUB + ((d >> 14) & (NSUB - 1))], 1u);
    }
  }
  __syncthreads();

  // R3: each rank -> sub-bin + residual2; then dedupe (bin,sub) pairs
  if (tid < NRANK) {
    int q = rk_q[tid];
    unsigned r1 = rk_res1[tid];
    unsigned cum = 0; int sub = NSUB - 1;
    for (int s2 = 0; s2 < NSUB; s2++) {
      unsigned h = hist2[q * NSUB + s2];
      if (cum + h > r1) { sub = s2; break; }
      cum += h;
    }
    rk_res2[tid] = r1 - cum;
    rk_pkey[tid] = (rk_bin[tid] << 7) | (unsigned)sub; // == dkey >> 14
  }
  __syncthreads();
  if (tid == 0) {
    int np = 0;
    for (int j = 0; j < NRANK; j++) {
      unsigned p = rk_pkey[j]; bool f = false;
      for (int k = 0; k < np; k++) if (plist[k] == p) { f = true; break; }
      if (!f) plist[np++] = p;
    }
    for (int a = 1; a < np; a++) {
      unsigned x = plist[a]; int k = a - 1;
      while (k >= 0 && plist[k] > x) { plist[k + 1] = plist[k]; k--; }
      plist[k + 1] = x;
    }
    s_np = np;
  }
  __syncthreads();
  if (tid < NRANK) rk_pslot[tid] = bfind(plist, s_np, rk_pkey[tid]);
  if (tid < NRANK) cursors[tid] = 0;
  __syncthreads();

  // R4: collect candidates for needed (bin,sub) pairs (one streaming pass)
  {
    const int np = s_np;
    for (int c = tid; c < V_COLS; c += BLOCK) {
      if (c == g) continue;
      unsigned d = dkey(t_row[c]);
      int p = bfind(plist, np, d >> 14);
      if (p >= 0) {
        unsigned pos = atomicAdd(&cursors[p], 1u);
        if (pos < CAP) sh_buf[p * CAP + pos] = (((unsigned long long)d) << 32) | (unsigned)c;
      }
    }
  }
  __syncthreads();

  // R5: per-rank in-list selection (smaller packed key == earlier rank;
  // low 32 bits carry the column index => stable tie-break by index).
  if (tid < NRANK) {
    int p = rk_pslot[tid];
    int n = (int)cursors[p]; if (n > CAP) n = CAP;
    unsigned r2 = rk_res2[tid];
    int found = -1;
    for (int i = 0; i < n; i++) {
      unsigned long long ei = sh_buf[p * CAP + i];
      int cnt = 0;
      for (int k = 0; k < n; k++) cnt += (sh_buf[p * CAP + k] < ei) ? 1 : 0;
      if (cnt == (int)r2) { found = (int)(ei & 0xFFFFFFFFull); break; }
    }
    if (found < 0) found = (n > 0) ? (int)(sh_buf[p * CAP] & 0xFFFFFFFFull) : 0;
    sel_idx[tid] = found;
  }
  __syncthreads();

  // ---------------- Final per-row loss (thread 0, precise math) ------------
  if (tid == 0) {
    const float lzt = logf(sumt), lzs = logf(sums);
    // tckd: 2-way CE between [p_t(gt), 1-p_t(gt)] and log of student pair
    float lpt = (sh_tgt - maxt) - lzt; float p_t = expf(lpt);
    float lps = (sh_sgt - maxs) - lzs; float p_s = expf(lps);
    float tckd = -(p_t * lps + (1.0f - p_t) * logf(fmaxf(1.0f - p_s, 1e-37f)));

    // pckd over 10 pos ranks
    float tp[NPOS], sp[NPOS];
    for (int j = 0; j < NPOS; j++) {
      int idx = sel_idx[j];
      tp[j] = t_row[idx];
      sp[j] = s_row[idx] + gumbel_of(u_row[idx]);
    }
    float mt2 = -INFINITY, ms2 = -INFINITY;
    for (int j = 0; j < NPOS; j++) { mt2 = fmaxf(mt2, tp[j]); ms2 = fmaxf(ms2, sp[j]); }
    float zt = 0.f, zs = 0.f;
    for (int j = 0; j < NPOS; j++) { zt += expf(tp[j] - mt2); zs += expf(sp[j] - ms2); }
    float lz2 = logf(zs);
    float pckd = 0.f;
    for (int j = 0; j < NPOS; j++) {
      float w = expf(tp[j] - mt2) / zt;
      pckd -= w * ((sp[j] - ms2) - lz2);
    }

    // nckd over 50 neg ranks
    float tn[NNEG], sn[NNEG];
    for (int j = 0; j < NNEG; j++) {
      int idx = sel_idx[NPOS + j];
      tn[j] = t_row[idx];
      sn[j] = s_row[idx] + gumbel_of(u_row[idx]);
    }
    float mt3 = -INFINITY, ms3 = -INFINITY;
    for (int j = 0; j < NNEG; j++) { mt3 = fmaxf(mt3, tn[j]); ms3 = fmaxf(ms3, sn[j]); }
    float zt3 = 0.f, zs3 = 0.f;
    for (int j = 0; j < NNEG; j++) { zt3 += expf(tn[j] - mt3); zs3 += expf(sn[j] - ms3); }
    float lz3 = logf(zs3);
    float nckd = 0.f;
    for (int j = 0; j < NNEG; j++) {
      float w = expf(tn[j] - mt3) / zt3;
      nckd -= w * ((sn[j] - ms3) - lz3);
    }

    row_loss[row] = (tckd + 1.0f * pckd + 8.0f * nckd) * (1.0f / (float)B_ROWS);
  }
}

__global__ void __launch_bounds__(BLOCK)
kd_reduce(const float* __restrict__ rl, float* __restrict__ out) {
  __shared__ float sh[16];
  int tid = threadIdx.x;
  float v = rl[tid]; // exactly B_ROWS == BLOCK entries
  for (int o = 16; o > 0; o >>= 1) v += __shfl_xor(v, o, 32);
  if ((tid & 31) == 0) sh[tid >> 5] = v;
  __syncthreads();
  if (tid == 0) { float a = 0.f; for (int i = 0; i < 16; i++) a += sh[i]; out[0] = a; }
}

extern "C" void kernel_launch(void* const* d_in, const int* in_sizes, int n_in,
                              void* d_out, int out_size, void* d_ws, size_t ws_size,
                              hipStream_t stream) {
  const float* t_score = (const float*)d_in[0];
  const float* s_score = (const float*)d_in[1];
  const int* gt = (const int*)d_in[2];
  const float* unif = (const float*)d_in[3];
  const int* neg_cols = (const int*)d_in[4];
  float* row_loss = (float*)d_ws; // 512 floats of scratch

  kd_main<<<B_ROWS, BLOCK, 0, stream>>>(t_score, s_score, gt, unif, neg_cols, row_loss);
  kd_reduce<<<1, BLOCK, 0, stream>>>(row_loss, (float*)d_out);
}